// ScaledDotProductAttention_71519795413418
// MI455X (gfx1250) — compile-verified
//
#include <hip/hip_runtime.h>

typedef __attribute__((ext_vector_type(2))) float v2f;
typedef __attribute__((ext_vector_type(4))) float v4f;
typedef __attribute__((ext_vector_type(4))) int   v4i;
typedef __attribute__((ext_vector_type(8))) float v8f;

// address-space qualified vector pointers for the async-to-LDS builtin
typedef __attribute__((address_space(1))) v4i gas_v4i;   // global
typedef __attribute__((address_space(3))) v4i las_v4i;   // LDS

#define B_      2
#define H_      12
#define S_      2048
#define D_      64
#define SCALE   0.125f      // 1/sqrt(64)
#define BAND    32          // rows of the attention band held in LDS
#define BSTRIDE 2052        // 2048 + 4 pad: row r shifts LDS banks by 4

// ---- gfx1250 async global->LDS staging (ASYNCcnt path), with fallback -----
#if defined(__has_builtin)
# if __has_builtin(__builtin_amdgcn_global_load_async_to_lds_b128)
#  define HAVE_ASYNC_LD 1
# endif
# if __has_builtin(__builtin_amdgcn_s_wait_asynccnt)
#  define HAVE_WAIT_ASYNC 1
# endif
#endif
#ifndef HAVE_ASYNC_LD
# define HAVE_ASYNC_LD 0
#endif
#ifndef HAVE_WAIT_ASYNC
# define HAVE_WAIT_ASYNC 0
#endif

__device__ __forceinline__ void stage16(const float* __restrict__ g, float* l) {
#if HAVE_ASYNC_LD
    // global_load_async_to_lds_b128: L2 -> LDS, no VGPR round trip
    __builtin_amdgcn_global_load_async_to_lds_b128(
        (gas_v4i*)g, (las_v4i*)l, 0, 0);
#else
    *(v4f*)l = *(const v4f*)g;          // load + ds_store fallback
#endif
}

__device__ __forceinline__ void stage_wait() {
#if HAVE_ASYNC_LD
# if HAVE_WAIT_ASYNC
    __builtin_amdgcn_s_wait_asynccnt(0);
# else
    asm volatile("s_wait_asynccnt 0x0" ::: "memory");
# endif
#endif
}

// ---------------------------------------------------------------------------
// Fully fused attention: one block owns a 32-row band of one (b,h) slice.
//   phase 1: band = scale * Q @ K^T   (WMMA f32 16x16x4, async K ping-pong)
//   phase 2: masked-fill (mask==0 -> 1e-9), row softmax in LDS
//   phase 4: out = band @ V, with the attention band streamed to HBM (NT)
//            chunk-by-chunk, overlapped with the WMMA stream.
// Block: 256 threads = 8 wave32, wave grid 2 (rows) x 4 (cols).
// LDS: 262,656 + 8,704 + 34,816 = 306,176 B (< 320 KB WGP budget).
// ---------------------------------------------------------------------------
__global__ __launch_bounds__(256)
void fused_attn_kernel(const float* __restrict__ q,
                       const float* __restrict__ k,
                       const float* __restrict__ v,
                       const int*   __restrict__ mask,
                       float*       __restrict__ out,
                       float*       __restrict__ attn) {
    __shared__ float band[BAND][BSTRIDE];
    __shared__ float Qs[BAND][68];
    __shared__ float KVs[2][64][68];        // ping-pong staging: K, then V

    const int t       = threadIdx.x;
    const int bh      = blockIdx.y;
    const int b       = bh / H_;
    const int rowBase = blockIdx.x * BAND;

    const int wave = t >> 5;
    const int lane = t & 31;
    const int half = lane >> 4;        // 0: lanes 0-15, 1: lanes 16-31
    const int r16  = lane & 15;
    const int wm   = wave & 1;         // 16-row subtile
    const int wn   = wave >> 1;        // 16-col subtile

    const float* Qg = q + ((size_t)bh * S_ + rowBase) * D_;   // 32x64 contiguous
    const float* Kg = k + (size_t)bh * S_ * D_;
    const float* Vg = v + (size_t)bh * S_ * D_;

    // ---- stage Q band (32x64) and K chunk 0, async ----
#pragma unroll
    for (int i = 0; i < 2; ++i) {
        int idx4 = t + i * 256;                 // 0..511
        int r = idx4 >> 4, c4 = idx4 & 15;
        stage16(Qg + (size_t)idx4 * 4, &Qs[r][c4 * 4]);
    }
#pragma unroll
    for (int i = 0; i < 4; ++i) {
        int idx4 = t + i * 256;                 // 0..1023
        int r = idx4 >> 4, c4 = idx4 & 15;
        stage16(Kg + (size_t)idx4 * 4, &KVs[0][r][c4 * 4]);
    }
    stage_wait();
    __syncthreads();

    // ================= phase 1: logits band = scale * Q K^T =================
    for (int c = 0; c < S_ / 64; ++c) {
        const int cur = c & 1;
        if (c + 1 < S_ / 64) {                  // async prefetch next K chunk
#pragma unroll
            for (int i = 0; i < 4; ++i) {
                int idx4 = t + i * 256;
                int r = idx4 >> 4, c4 = idx4 & 15;
                stage16(Kg + (size_t)(c + 1) * 64 * D_ + (size_t)idx4 * 4,
                        &KVs[1 - cur][r][c4 * 4]);
            }
        }
        v8f acc = {0.f, 0.f, 0.f, 0.f, 0.f, 0.f, 0.f, 0.f};
#pragma unroll
        for (int kk = 0; kk < D_; kk += 4) {
            v2f a, bf;
            a.x  = Qs[wm * 16 + r16][kk + half * 2 + 0];
            a.y  = Qs[wm * 16 + r16][kk + half * 2 + 1];
            bf.x = KVs[cur][wn * 16 + r16][kk + half * 2 + 0];
            bf.y = KVs[cur][wn * 16 + r16][kk + half * 2 + 1];
            acc = __builtin_amdgcn_wmma_f32_16x16x4_f32(false, a, false, bf,
                                                        (short)0, acc, false, false);
        }
        const int colL = c * 64 + wn * 16 + r16;
#pragma unroll
        for (int e = 0; e < 8; ++e)
            band[wm * 16 + half * 8 + e][colL] = acc[e] * SCALE;
        stage_wait();
        __syncthreads();
    }

    // ================= phase 2a: masked fill (mask==0 -> 1e-9) ==============
    const int* Mg = mask + (size_t)b * S_ * S_ + (size_t)rowBase * S_;
#pragma unroll 4
    for (int i = 0; i < 64; ++i) {
        int idx4 = t + i * 256;                 // 0..16383
        int r = idx4 >> 9, c4 = idx4 & 511;     // 512 float4 per row
        v4i mv = ((const v4i*)(Mg + (size_t)r * S_))[c4];
        v4f bv = *(v4f*)&band[r][c4 * 4];
        bv.x = (mv.x == 0) ? 1e-9f : bv.x;
        bv.y = (mv.y == 0) ? 1e-9f : bv.y;
        bv.z = (mv.z == 0) ? 1e-9f : bv.z;
        bv.w = (mv.w == 0) ? 1e-9f : bv.w;
        *(v4f*)&band[r][c4 * 4] = bv;
    }
    __syncthreads();

    // ================= phase 2b: row softmax in LDS (b128 passes) ===========
    // wave w owns rows 4w .. 4w+3.
    for (int rr = 0; rr < 4; ++rr) {
        const int row = wave * 4 + rr;
        float m = -3.402823466e38f;
        for (int i4 = lane; i4 < S_ / 4; i4 += 32) {
            v4f x = *(const v4f*)&band[row][i4 * 4];
            m = fmaxf(m, fmaxf(fmaxf(x.x, x.y), fmaxf(x.z, x.w)));
        }
        for (int off = 16; off; off >>= 1) m = fmaxf(m, __shfl_xor(m, off));

        float s = 0.f;
        for (int i4 = lane; i4 < S_ / 4; i4 += 32) {
            v4f x = *(v4f*)&band[row][i4 * 4];
            x.x = __expf(x.x - m); x.y = __expf(x.y - m);
            x.z = __expf(x.z - m); x.w = __expf(x.w - m);
            *(v4f*)&band[row][i4 * 4] = x;
            s += (x.x + x.y) + (x.z + x.w);
        }
        for (int off = 16; off; off >>= 1) s += __shfl_xor(s, off);
        const float inv = 1.0f / s;
        for (int i4 = lane; i4 < S_ / 4; i4 += 32) {
            v4f x = *(v4f*)&band[row][i4 * 4];
            x *= inv;
            *(v4f*)&band[row][i4 * 4] = x;
        }
    }
    __syncthreads();

    // ================= phase 4: out = band @ V, + streamed attn writeout ====
    // stage V chunk 0 (KVs free since phase 1 ended)
#pragma unroll
    for (int i = 0; i < 4; ++i) {
        int idx4 = t + i * 256;
        int r = idx4 >> 4, c4 = idx4 & 15;
        stage16(Vg + (size_t)idx4 * 4, &KVs[0][r][c4 * 4]);
    }
    stage_wait();
    __syncthreads();

    float* Ag = attn + ((size_t)bh * S_ + rowBase) * S_;

    v8f acc = {0.f, 0.f, 0.f, 0.f, 0.f, 0.f, 0.f, 0.f};
    for (int c = 0; c < S_ / 64; ++c) {
        const int cur = c & 1;
        if (c + 1 < S_ / 64) {                  // async prefetch next V chunk
#pragma unroll
            for (int i = 0; i < 4; ++i) {
                int idx4 = t + i * 256;
                int r = idx4 >> 4, c4 = idx4 & 15;
                stage16(Vg + (size_t)(c + 1) * 64 * D_ + (size_t)idx4 * 4,
                        &KVs[1 - cur][r][c4 * 4]);
            }
        }
#pragma unroll
        for (int kk = 0; kk < 64; kk += 4) {
            v2f a, bf;
            const int j = c * 64 + kk;
            a.x  = band[wm * 16 + r16][j + half * 2 + 0];
            a.y  = band[wm * 16 + r16][j + half * 2 + 1];
            bf.x = KVs[cur][kk + half * 2 + 0][wn * 16 + r16];
            bf.y = KVs[cur][kk + half * 2 + 1][wn * 16 + r16];
            acc = __builtin_amdgcn_wmma_f32_16x16x4_f32(false, a, false, bf,
                                                        (short)0, acc, false, false);
        }
        // stream attention columns [c*64, c*64+64) to HBM (non-temporal),
        // overlapped with the WMMA stream; uses STOREcnt, not ASYNCcnt.
#pragma unroll
        for (int i = 0; i < 2; ++i) {
            int idx4 = t + i * 256;             // 0..511
            int r = idx4 >> 4, c4 = idx4 & 15;  // 16 float4 per 64-col row slice
            v4f bv = *(const v4f*)&band[r][c * 64 + c4 * 4];
            __builtin_nontemporal_store(bv, (v4f*)(Ag + (size_t)r * S_ + c * 64) + c4);
        }
        stage_wait();
        __syncthreads();
    }

    const int col = wn * 16 + r16;
#pragma unroll
    for (int e = 0; e < 8; ++e) {
        int row = rowBase + wm * 16 + half * 8 + e;
        out[((size_t)bh * S_ + row) * D_ + col] = acc[e];
    }
}

// ---------------------------------------------------------------------------
extern "C" void kernel_launch(void* const* d_in, const int* in_sizes, int n_in,
                              void* d_out, int out_size, void* d_ws, size_t ws_size,
                              hipStream_t stream) {
    (void)in_sizes; (void)n_in; (void)out_size; (void)d_ws; (void)ws_size;

    const float* q    = (const float*)d_in[0];
    const float* k    = (const float*)d_in[1];
    const float* v    = (const float*)d_in[2];
    const int*   mask = (const int*)d_in[3];

    float* out  = (float*)d_out;                           // [B,H,S,D]
    float* attn = out + (size_t)B_ * H_ * S_ * D_;         // [B,H,S,S]

    dim3 grid(S_ / BAND, B_ * H_);
    fused_attn_kernel<<<grid, 256, 0, stream>>>(q, k, v, mask, out, attn);
}